// Ocean_e2e_51874615001400
// MI455X (gfx1250) — compile-verified
//
#include <hip/hip_runtime.h>
#include <stdint.h>

// Ocean advection-diffusion, 48 explicit steps, MI455X (gfx1250, wave32).
// Per step: fused (derivatives -> advective tendency -> 3x3 binomial smooth -> mask).
// One wave = one 16x16 output tile. T halo staged via async global->LDS;
// vertical smoothing pass done with v_wmma_f32_16x16x4_f32 (5 chained K=4 chunks
// over an 18-row + 2-pad banded weight matrix).

typedef float v2f __attribute__((ext_vector_type(2)));
typedef float v8f __attribute__((ext_vector_type(8)));

#define RE      6371000.0f
#define D2R     0.017453292519943295f
#define DTSTEP  600.0f
#define HH      1024
#define WW      1024
#define NB      8
#define NSTEPS  48

__global__ __launch_bounds__(256) void ocean_step_kernel(
    const float* __restrict__ Tin, const float* __restrict__ ug,
    const float* __restrict__ vg,  const float* __restrict__ lat,
    const float* __restrict__ lon, const float* __restrict__ mask,
    float* __restrict__ Tout)
{
    // per-wave LDS regions (8 waves / block)
    __shared__ float sT[8 * 400];   // 20x20 T tile (rows/cols -2..17)
    __shared__ float sU[8 * 324];   // 18x18 U = T + dt*F (rows/cols -1..16)
    __shared__ float sH[8 * 320];   // 20x16 horizontal-smoothed (rows -1..16 + 2 zero pad)

    const int lane = threadIdx.x & 31;
    const int wv   = threadIdx.x >> 5;
    const int tile = blockIdx.x * 8 + wv;          // 32768 tiles total
    const int b    = tile >> 12;                   // / (64*64)
    const int rem  = tile & 4095;
    const int ty0  = (rem >> 6) << 4;              // tile row origin
    const int tx0  = (rem & 63) << 4;              // tile col origin

    const float dlat   = lat[1] - lat[0];
    const float dlon   = lon[1] - lon[0];
    const float dy     = RE * D2R * dlat;
    const float dxc    = RE * D2R * dlon;
    const float inv_dy  = 1.0f / dy;
    const float inv_2dy = 0.5f * inv_dy;

    const size_t plane = (size_t)b * ((size_t)HH * WW);
    const float* Tb = Tin + plane;

    float* myT = &sT[wv * 400];
    float* myU = &sU[wv * 324];
    float* myH = &sH[wv * 320];

    // ---- Phase 1: async-load 20x20 T halo tile into LDS (clamped addresses;
    //      out-of-domain values are never consumed) ----
    for (int i = lane; i < 400; i += 32) {
        int r = i / 20, c = i - r * 20;
        int y = ty0 - 2 + r; y = y < 0 ? 0 : (y > HH - 1 ? HH - 1 : y);
        int x = tx0 - 2 + c; x = x < 0 ? 0 : (x > WW - 1 ? WW - 1 : x);
        uint64_t gaddr = (uint64_t)(uintptr_t)(Tb + (size_t)y * WW + x);
        uint32_t laddr = (uint32_t)(uintptr_t)&myT[i];
        asm volatile("global_load_async_to_lds_b32 %0, %1, off"
                     :: "v"(laddr), "v"(gaddr) : "memory");
    }
    asm volatile("s_wait_asynccnt 0" ::: "memory");
    __syncthreads();

    // ---- Phase 2: U = T + dt * ( -(ug*dT/dx + vg*dT/dy) * mask ) on 18x18,
    //      zero outside the global domain (== conv zero padding) ----
    for (int i = lane; i < 324; i += 32) {
        int r = i / 18, c = i - r * 18;
        int y = ty0 - 1 + r;
        int x = tx0 - 1 + c;
        float u = 0.0f;
        if (y >= 0 && y < HH && x >= 0 && x < WW) {
            int tr = r + 1, tc = c + 1;                 // T-tile coords of (y,x)
            float tc0 = myT[tr * 20 + tc];
            float dTdy;
            if (y == 0)            dTdy = (myT[(tr + 1) * 20 + tc] - tc0) * inv_dy;
            else if (y == HH - 1)  dTdy = (tc0 - myT[(tr - 1) * 20 + tc]) * inv_dy;
            else                   dTdy = (myT[(tr + 1) * 20 + tc] - myT[(tr - 1) * 20 + tc]) * inv_2dy;
            float inv_dx = 1.0f / (dxc * cosf(lat[y] * D2R));
            float dTdx;
            if (x == 0)            dTdx = (myT[tr * 20 + tc + 1] - tc0) * inv_dx;
            else if (x == WW - 1)  dTdx = (tc0 - myT[tr * 20 + tc - 1]) * inv_dx;
            else                   dTdx = (myT[tr * 20 + tc + 1] - myT[tr * 20 + tc - 1]) * (0.5f * inv_dx);
            size_t gi = plane + (size_t)y * WW + x;
            float F = -(ug[gi] * dTdx + vg[gi] * dTdy) * mask[(size_t)y * WW + x];
            u = tc0 + DTSTEP * F;
        }
        myU[i] = u;
    }
    __syncthreads();

    // ---- Phase 3: horizontal [1,2,1]/4 pass -> H (rows 0..17 = U rows -1..16,
    //      cols 0..15; rows 18..19 zero-padded for the K=20 WMMA sweep) ----
    for (int i = lane; i < 320; i += 32) {
        int r = i >> 4, c = i & 15;
        float h = 0.0f;
        if (r < 18) {
            const float* Ur = &myU[r * 18];
            h = 0.25f * (Ur[c] + Ur[c + 2]) + 0.5f * Ur[c + 1];
        }
        myH[i] = h;
    }
    __syncthreads();

    // ---- Phase 4: vertical [1,2,1]/4 pass as D = A(16x20) * H(20x16) using
    //      five v_wmma_f32_16x16x4_f32. A[m,k] = {.25,.5,.25} for k = m,m+1,m+2.
    //      f32 A-frag layout: lane->M=lane%16; VGPR v + lane-half -> K = 2*(lane/16)+v.
    //      f32 B-frag layout: lane->N=lane%16; VGPR v + lane-half -> K = 2*(lane/16)+v. ----
    v8f acc = {0.f, 0.f, 0.f, 0.f, 0.f, 0.f, 0.f, 0.f};
    const int m   = lane & 15;          // also N for the B fragment / output column
    const int hi2 = (lane >> 4) << 1;   // 0 or 2
#pragma unroll
    for (int j0 = 0; j0 < 20; j0 += 4) {
        int k0 = j0 + hi2;
        int k1 = k0 + 1;
        v2f a, bf;
        a.x = (k0 == m) ? 0.25f : (k0 == m + 1) ? 0.5f : (k0 == m + 2) ? 0.25f : 0.0f;
        a.y = (k1 == m) ? 0.25f : (k1 == m + 1) ? 0.5f : (k1 == m + 2) ? 0.25f : 0.0f;
        bf.x = myH[k0 * 16 + m];
        bf.y = myH[k1 * 16 + m];
        acc = __builtin_amdgcn_wmma_f32_16x16x4_f32(
            /*neg_a=*/false, a, /*neg_b=*/false, bf,
            /*c_mod=*/(short)0, acc, /*reuse_a=*/false, /*reuse_b=*/false);
    }

    // ---- Phase 5: mask and store. C/D layout: VGPR i -> M = i + 8*(lane/16), N = lane%16.
    const int mbase = (lane >> 4) << 3;
    float* Ob = Tout + plane;
#pragma unroll
    for (int i = 0; i < 8; ++i) {
        int y = ty0 + mbase + i;
        int x = tx0 + m;
        Ob[(size_t)y * WW + x] = acc[i] * mask[(size_t)y * WW + x];
    }
}

extern "C" void kernel_launch(void* const* d_in, const int* in_sizes, int n_in,
                              void* d_out, int out_size, void* d_ws, size_t ws_size,
                              hipStream_t stream) {
    (void)in_sizes; (void)n_in; (void)out_size; (void)ws_size;
    const float* T0   = (const float*)d_in[0];
    const float* ug   = (const float*)d_in[1];
    const float* vg   = (const float*)d_in[2];
    const float* lat  = (const float*)d_in[3];
    const float* lon  = (const float*)d_in[4];
    const float* mask = (const float*)d_in[5];

    float* bufA = (float*)d_ws;    // ping (needs 8*1024*1024*4 = 33.6 MB of ws)
    float* bufB = (float*)d_out;   // pong; step 47 (odd) lands here

    const int nTiles = NB * 64 * 64;       // 32768
    dim3 grid(nTiles / 8), block(256);

    const float* src = T0;
    for (int s = 0; s < NSTEPS; ++s) {
        float* dst = (s & 1) ? bufB : bufA;
        ocean_step_kernel<<<grid, block, 0, stream>>>(src, ug, vg, lat, lon, mask, dst);
        src = dst;
    }
}